// PointPillarsScatter_18468359373358
// MI455X (gfx1250) — compile-verified
//
#include <hip/hip_runtime.h>
#include <stdint.h>
#include <stddef.h>

// PointPillars scatter-max for MI455X (gfx1250, wave32).
// Memory-bound: 256MB output write floor ~= 11us @ 23.3 TB/s.
// Strategy: compact touched cells into an L2-resident (B,M,C) canvas via
// claim-map + u32 atomic max on order-preserving float keys, then one
// streaming non-temporal pass that writes every output element exactly once.
// Feature tiles are staged global->LDS with gfx1250 async-LDS loads
// (ASYNCcnt) overlapped with the claim CAS round-trip.

#define ZDIM 512
#define XDIM 512
#define CDIM 64

typedef float    v4f __attribute__((ext_vector_type(4)));
typedef unsigned v4u __attribute__((ext_vector_type(4)));

// Monotone float->u32 key with key(-inf) == 0 and key order == float order.
__device__ __forceinline__ unsigned ordkey(float f) {
    unsigned u = __float_as_uint(f);
    u = (u & 0x80000000u) ? ~u : (u | 0x80000000u);
    return u - 0x007FFFFFu;   // ord(-inf) = ~0xFF800000 = 0x007FFFFF
}
__device__ __forceinline__ float keydec(unsigned k) {
    unsigned u = k + 0x007FFFFFu;
    u = (u & 0x80000000u) ? (u & 0x7FFFFFFFu) : ~u;
    return __uint_as_float(u);
}

// ---------------- fast path (workspace big enough) ----------------

__global__ void zero_ws_kernel(v4u* __restrict__ p, size_t n4) {
    size_t i = (size_t)blockIdx.x * blockDim.x + threadIdx.x;
    if (i < n4) {
        v4u z = (v4u)0u;
        p[i] = z;   // small (16MB), keep resident in L2
    }
}

// 4 points per 256-thread block; 64 lanes per point (one per channel).
// Feature tile (1KB, contiguous) staged via async global->LDS copy while the
// per-point claim CAS is in flight.
__global__ void scatter_compact_kernel(const float* __restrict__ feat,
                                       const int*   __restrict__ coords,
                                       unsigned*    __restrict__ claim,
                                       unsigned*    __restrict__ compact,
                                       int M, int npts) {
    __shared__ float    tile[256];   // 4 points * 64 channels
    __shared__ unsigned srow[4];

    const int tid = threadIdx.x;
    const int p   = tid >> 6;          // point slot within block
    const int c   = tid & 63;          // channel
    const int pt  = blockIdx.x * 4 + p;
    const bool ok = (pt < npts);

    // Kick off the async feature fetch: feat[blockIdx*256 + tid] -> tile[tid].
    // Tracked by ASYNCcnt; each lane later reads back only its own slot, so a
    // per-wave s_wait_asynccnt 0 is sufficient ordering.
    if (ok) {
        unsigned lds_addr = (unsigned)(uintptr_t)(&tile[tid]);
        const float* gp = feat + ((size_t)blockIdx.x * 256 + tid);
        asm volatile("global_load_async_to_lds_b32 %0, %1, off"
                     :: "v"(lds_addr), "v"(gp)
                     : "memory");
    }

    // Overlap: allocate the compact row while the async copy is in flight.
    if (ok && c == 0) {
        const int b = pt / M;
        const int m = pt - b * M;
        const int z = coords[pt * 3 + 0] & (ZDIM - 1);
        const int x = coords[pt * 3 + 2] & (XDIM - 1);
        const unsigned cell = ((unsigned)b << 18) | ((unsigned)z << 9) | (unsigned)x;
        // Single CAS allocates the compact row: winner uses its own m,
        // losers read the winner's m from the returned old value.
        const unsigned old = atomicCAS(&claim[cell], 0u, (unsigned)(m + 1));
        srow[p] = (old == 0u) ? (unsigned)m : (old - 1u);
    }
    __syncthreads();

    // Drain this wave's async copies before touching the tile.
    asm volatile("s_wait_asynccnt 0x0" ::: "memory");

    if (ok) {
        const int b = pt / M;
        const unsigned row = srow[p];
        const float f = tile[tid];
        const size_t idx = ((size_t)b * (size_t)M + row) * CDIM + c; // dense row
        atomicMax(&compact[idx], ordkey(f));                    // global_atomic_max_u32
    }
}

// One thread -> 4 consecutive x values -> one 128-bit NT store.
__global__ void write_out_kernel(const unsigned* __restrict__ claim,
                                 const unsigned* __restrict__ compact,
                                 float* __restrict__ out,
                                 int M, size_t n4) {
    const size_t i4 = (size_t)blockIdx.x * blockDim.x + threadIdx.x;
    if (i4 >= n4) return;
    const size_t base = i4 * 4;                       // ((b*C+c)*Z+z)*X+x
    const unsigned x = (unsigned)(base & (XDIM - 1));
    const unsigned z = (unsigned)((base >> 9)  & (ZDIM - 1));
    const unsigned c = (unsigned)((base >> 18) & (CDIM - 1));
    const unsigned b = (unsigned)(base >> 24);

    const v4u cl = *(const v4u*)(claim + (((size_t)b << 18) | ((size_t)z << 9) | x));

    v4f o;
    #pragma unroll
    for (int j = 0; j < 4; ++j) {
        const unsigned cj = cl[j];
        float f = 0.0f;
        if (cj != 0u) {
            const unsigned row = cj - 1u;
            const unsigned key = compact[((size_t)b * (size_t)M + row) * CDIM + c];
            if (key != 0u) f = keydec(key);
        }
        o[j] = f;
    }
    __builtin_nontemporal_store(o, (v4f*)(out + base));   // global_store_b128 NT
}

// ---------------- fallback path (tiny workspace) ----------------

__global__ void fb_init_kernel(v4u* __restrict__ outk, size_t n4) {
    size_t i = (size_t)blockIdx.x * blockDim.x + threadIdx.x;
    if (i < n4) {
        v4u z = (v4u)0u;                               // key 0 == -inf sentinel == 0.0f bits
        __builtin_nontemporal_store(z, &outk[i]);
    }
}

__global__ void fb_scatter_kernel(const float* __restrict__ feat,
                                  const int*   __restrict__ coords,
                                  unsigned*    __restrict__ outk,
                                  int M, int total) {
    const int t = blockIdx.x * blockDim.x + threadIdx.x;
    if (t >= total) return;
    const int c  = t & 63;
    const int pt = t >> 6;
    const int b  = pt / M;
    const int z  = coords[pt * 3 + 0] & (ZDIM - 1);
    const int x  = coords[pt * 3 + 2] & (XDIM - 1);
    const size_t idx = ((((size_t)b << 6) | (size_t)c) << 18)
                     | ((size_t)z << 9) | (size_t)x;   // ((b*64+c)*512+z)*512+x
    atomicMax(&outk[idx], ordkey(feat[(size_t)pt * CDIM + c]));
}

__global__ void fb_decode_kernel(unsigned* __restrict__ outk, size_t n) {
    const size_t i = (size_t)blockIdx.x * blockDim.x + threadIdx.x;
    if (i >= n) return;
    const unsigned k = __builtin_nontemporal_load(&outk[i]);
    if (k != 0u) outk[i] = __float_as_uint(keydec(k));  // empty cells already 0.0f bits
}

// ---------------- host launch ----------------

extern "C" void kernel_launch(void* const* d_in, const int* in_sizes, int n_in,
                              void* d_out, int out_size, void* d_ws, size_t ws_size,
                              hipStream_t stream) {
    (void)n_in;
    const float* feat   = (const float*)d_in[0];
    const int*   coords = (const int*)d_in[1];
    float*       out    = (float*)d_out;

    const size_t outN = (size_t)out_size;        // B * 64 * 512 * 512
    const int    B    = (int)(outN >> 24);
    const int    npts = in_sizes[1] / 3;         // B * M
    const int    M    = npts / (B > 0 ? B : 1);

    const size_t claimWords   = (size_t)B << 18;            // B*512*512
    const size_t compactWords = (size_t)B * (size_t)M * CDIM;
    const size_t needBytes    = (claimWords + compactWords) * sizeof(unsigned);

    if (ws_size >= needBytes) {
        unsigned* claim   = (unsigned*)d_ws;
        unsigned* compact = claim + claimWords;

        const size_t zw4 = (claimWords + compactWords + 3) / 4;
        zero_ws_kernel<<<(unsigned)((zw4 + 255) / 256), 256, 0, stream>>>((v4u*)d_ws, zw4);

        const int sblocks = (npts + 3) / 4;
        scatter_compact_kernel<<<sblocks, 256, 0, stream>>>(feat, coords, claim, compact, M, npts);

        const size_t n4 = outN / 4;
        write_out_kernel<<<(unsigned)((n4 + 255) / 256), 256, 0, stream>>>(claim, compact, out, M, n4);
    } else {
        const size_t n4 = outN / 4;
        fb_init_kernel<<<(unsigned)((n4 + 255) / 256), 256, 0, stream>>>((v4u*)out, n4);

        const int total = npts * CDIM;
        fb_scatter_kernel<<<(total + 255) / 256, 256, 0, stream>>>(feat, coords, (unsigned*)out, M, total);

        fb_decode_kernel<<<(unsigned)((outN + 255) / 256), 256, 0, stream>>>((unsigned*)out, outN);
    }
}